// LGNNodeLevel_65481071394896
// MI455X (gfx1250) — compile-verified
//
#include <hip/hip_runtime.h>

// ---------------------------------------------------------------------------
// LGN node-level equivariant layer for MI455X (gfx1250), wave32 + WMMA f32.
//
// Dominant work: tmp_{l1,l2}[(i,s,y),(t,x)] = Edge_{l2} x MaskedNode_{l1}
// fused over all l1 (B matrix = 144 cols = 9 WMMA tiles) so the 75MB of edge
// data is streamed exactly once (HBM floor ~3.2us @ 23.3 TB/s).
// B chunks are staged into LDS via the Tensor Data Mover (TENSORcnt) when the
// toolchain exposes the builtin; the final CatMix is a second WMMA GEMM.
//
// Workspace requirement: ~60.5 MB (guarded).
// ---------------------------------------------------------------------------

typedef __attribute__((ext_vector_type(2))) float    v2f;
typedef __attribute__((ext_vector_type(8))) float    v8f;
typedef __attribute__((ext_vector_type(4))) unsigned u32x4;
typedef __attribute__((ext_vector_type(8))) int      i32x8;
typedef __attribute__((ext_vector_type(4))) int      i32x4;

#if __has_builtin(__builtin_amdgcn_tensor_load_to_lds) && \
    __has_builtin(__builtin_amdgcn_s_wait_tensorcnt)
#define HAVE_TDM 1
#else
#define HAVE_TDM 0
#endif

#define BB 8
#define NN 256
#define TT 16
#define SS 4
#define NCH 16

// 15 (l, pair) entries in reference order:
// l=0: (0,0)(1,1)(2,2)   l=1: (0,1)(1,0)(1,1)(1,2)(2,1)(2,2)
// l=2: (0,2)(1,1)(1,2)(2,0)(2,1)(2,2)
__constant__ int c_entL1[15] = {0,1,2, 0,1,1,1,2,2, 0,1,1,2,2,2};
__constant__ int c_entL2[15] = {0,1,2, 1,0,1,2,1,2, 2,1,2,0,1,2};
__constant__ int c_cgOff[15] = {0,1,10, 35,44,53,80,125,170, 245,270,315,390,415,490}; // 615 total
__constant__ int c_entBase[3] = {0,3,9};
__constant__ int c_npair[3]   = {3,6,6};
// tmp_{l1,l2} offsets (q = l1*3+l2), sizes 16384*d1*d2 floats (per batch)
__constant__ int c_tmpOffQ[9] = {0,16384,65536,147456,196608,344064,589824,671744,917504};
#define TMP_PER_B 1327104
// fused-GEMM tile metadata: 9 column tiles over combined 144-col B matrix
__constant__ int c_colBase[9]      = {0,16,32,48,64,80,96,112,128};
__constant__ int c_tileL1[9]       = {0,1,1,1,2,2,2,2,2};
__constant__ int c_tileColLocal[9] = {0,0,16,32,0,16,32,48,64};

// ---------------------------------------------------------------------------
// Tensor Data Mover: contiguous 1-D tile (nElems x 4B) global -> LDS.
// D# per CDNA5 ISA 8.3/8.4: group0 = {count=1, lds_addr, global_addr, type=2},
// group1 = {data_size=4B, tensor_dim0=tile_dim0=nElems, tensor_dim1=1}.
// ---------------------------------------------------------------------------
__device__ __forceinline__ void tdm_load_chunk(const float* gsrc, unsigned ldsByteOff,
                                               unsigned nElems) {
#if HAVE_TDM
    unsigned long long ga = (unsigned long long)(size_t)gsrc;
    u32x4 g0;
    g0[0] = 1u;                                            // count=1, user desc
    g0[1] = ldsByteOff;                                    // lds_addr (bytes)
    g0[2] = (unsigned)(ga & 0xffffffffu);                  // global_addr[31:0]
    g0[3] = (unsigned)((ga >> 32) & 0x01ffffffu) | (2u << 30); // addr[56:32] | type=2
    i32x8 g1;
    g1[0] = (int)(2u << 16);                               // data_size = 2 -> 4 bytes
    g1[1] = (int)((nElems & 0xffffu) << 16);               // tensor_dim0[15:0]
    g1[2] = (int)((nElems >> 16) & 0xffffu) | (1 << 16);   // tensor_dim0[31:16] | tensor_dim1=1
    g1[3] = (int)((nElems & 0xffffu) << 16);               // tile_dim0 (tensor_dim1 hi = 0)
    g1[4] = 0; g1[5] = 0; g1[6] = 0; g1[7] = 0;            // no dim2, no strides
    i32x4 z4 = {0, 0, 0, 0};
#if __clang_major__ >= 23
    i32x8 z8 = {0, 0, 0, 0, 0, 0, 0, 0};
    __builtin_amdgcn_tensor_load_to_lds(g0, g1, z4, z4, z8, 0);
#else
    __builtin_amdgcn_tensor_load_to_lds(g0, g1, z4, z4, 0);
#endif
    __builtin_amdgcn_s_wait_tensorcnt(0);
#else
    (void)gsrc; (void)ldsByteOff; (void)nElems;
#endif
}

// ---------------------------------------------------------------------------
// CG tables via Racah formula, computed on device (once per launch).
// ---------------------------------------------------------------------------
__device__ double dfact(int n) {
    double r = 1.0;
    for (int i = 2; i <= n; ++i) r *= (double)i;
    return r;
}

__device__ float cg_coeff(int j1, int m1, int j2, int m2, int j, int m) {
    if (m1 + m2 != m) return 0.0f;
    double pre = sqrt((double)(2 * j + 1) * dfact(j + j1 - j2) * dfact(j - j1 + j2) *
                      dfact(j1 + j2 - j) / dfact(j1 + j2 + j + 1));
    pre *= sqrt(dfact(j + m) * dfact(j - m) * dfact(j1 - m1) * dfact(j1 + m1) *
                dfact(j2 - m2) * dfact(j2 + m2));
    int k0 = 0;
    if (j2 - j - m1 > k0) k0 = j2 - j - m1;
    if (j1 + m2 - j > k0) k0 = j1 + m2 - j;
    int k1 = j1 + j2 - j;
    if (j1 - m1 < k1) k1 = j1 - m1;
    if (j2 + m2 < k1) k1 = j2 + m2;
    double s = 0.0;
    for (int k = k0; k <= k1; ++k) {
        double den = dfact(k) * dfact(j1 + j2 - j - k) * dfact(j1 - m1 - k) *
                     dfact(j2 + m2 - k) * dfact(j - j2 + m1 + k) * dfact(j - j1 - m2 + k);
        s += ((k & 1) ? -1.0 : 1.0) / den;
    }
    return (float)(pre * s);
}

__global__ void cg_kernel(float* __restrict__ cgw) {
    int idx = blockIdx.x * blockDim.x + threadIdx.x;
    if (idx >= 615) return;
    int e = 14;
    for (int t = 0; t < 15; ++t) {
        int lt = (t < 3) ? 0 : ((t < 9) ? 1 : 2);
        int sz = (2 * c_entL1[t] + 1) * (2 * c_entL2[t] + 1) * (2 * lt + 1);
        if (idx < c_cgOff[t] + sz) { e = t; break; }
    }
    int l  = (e < 3) ? 0 : ((e < 9) ? 1 : 2);
    int l1 = c_entL1[e], l2 = c_entL2[e];
    int dl = 2 * l + 1, d2 = 2 * l2 + 1;
    int local = idx - c_cgOff[e];
    int im = local % dl;
    int rest = local / dl;
    int y = rest % d2;
    int x = rest / d2;
    cgw[idx] = cg_coeff(l1, x - l1, l2, y - l2, l, im - l);
}

// ---------------------------------------------------------------------------
// Pack masked nodes into combined B matrix: nm[b][j][c], c in [0,144):
//   c<16 -> l1=0 ; 16..63 -> l1=1 (t*3+x) ; 64..143 -> l1=2 (t*5+x)
// ---------------------------------------------------------------------------
__global__ void mask_node_kernel(const float* __restrict__ n0, const float* __restrict__ n1,
                                 const float* __restrict__ n2, const int* __restrict__ mask,
                                 float* __restrict__ nm) {
    int b = blockIdx.y;
    int idx = blockIdx.x * 256 + threadIdx.x;   // < 36864
    int j = idx / 144, c = idx % 144;
    float mf = (float)mask[b * NN + j];
    float v;
    if (c < 16) {
        v = n0[(size_t)(b * NN + j) * TT + c];
    } else if (c < 64) {
        int cc = c - 16;
        v = n1[((size_t)(b * NN + j) * TT + cc / 3) * 3 + cc % 3];
    } else {
        int cc = c - 64;
        v = n2[((size_t)(b * NN + j) * TT + cc / 5) * 5 + cc % 5];
    }
    nm[(size_t)b * 36864 + idx] = v * mf;
}

// ---------------------------------------------------------------------------
// Fused aggregate GEMM, one launch per l2, grid = (8*d2, B), block = 256 (8 waves).
// C[(i,s,y),(t,x)] over all l1 at once: 9 WMMA col tiles share one A fragment.
// A (edge) streamed from global (read exactly once, prefetched one chunk
// ahead), B staged in LDS chunks via TDM (fallback: cooperative copy).
// ---------------------------------------------------------------------------
__launch_bounds__(256)
__global__ void agg_gemm_kernel(const float* __restrict__ edge, const float* __restrict__ nmAll,
                                float* __restrict__ tmpAll, int l2) {
    const int d2 = 2 * l2 + 1;
    const int b = blockIdx.y;
    const int tid = threadIdx.x;
    const int wave = tid >> 5, lane = tid & 31;
    const int laneM = lane & 15, hi = lane >> 4;
    const int rowBase = blockIdx.x * 128 + wave * 16;

    const float* nm = nmAll + (size_t)b * 36864;
    const float* eb = edge + (size_t)b * NN * NN * SS * d2;
    const int jStride = SS * d2;            // stride of j in edge
    const int iStride = NN * SS * d2;       // stride of i in edge

    __shared__ float Bs[64 * 144];          // 36 KB K-chunk of B

    const v8f vzero = {0, 0, 0, 0, 0, 0, 0, 0};
    v8f acc[9];
#pragma unroll
    for (int t = 0; t < 9; ++t) acc[t] = vzero;

    // A-fragment base address for this lane (row = rowBase + laneM)
    const int r = rowBase + laneM;
    const int i = r / jStride;
    const int rem = r - i * jStride;        // s*d2 + y
    const float* ap = eb + (size_t)i * iStride + rem;

    for (int k0 = 0; k0 < NN; k0 += 64) {
        __syncthreads();
#if HAVE_TDM
        if (wave == 0)
            tdm_load_chunk(nm + (size_t)k0 * 144, (unsigned)(size_t)(&Bs[0]), 64 * 144);
#else
        for (int idx = tid; idx < 64 * 144; idx += 256)
            Bs[idx] = nm[(size_t)k0 * 144 + idx];
#endif
        __syncthreads();
        if (k0 + 64 < NN) {
            // stream the next edge K-chunk ahead of the WMMA chain
            __builtin_prefetch(ap + (size_t)(k0 + 64 + hi * 2) * jStride, 0, 1);
        }
        for (int kk = 0; kk < 64; kk += 4) {
            const int kb = kk + hi * 2;     // lane's K pair within the 4-step
            v2f a;
            a.x = ap[(size_t)(k0 + kb) * jStride];
            a.y = ap[(size_t)(k0 + kb + 1) * jStride];
            // preload all 9 B fragments, then issue the WMMA chain: batches the
            // ds_loads so their latency hides under the matrix pipe
            v2f bf[9];
#pragma unroll
            for (int ti = 0; ti < 9; ++ti) {
                bf[ti].x = Bs[kb * 144 + c_colBase[ti] + laneM];
                bf[ti].y = Bs[(kb + 1) * 144 + c_colBase[ti] + laneM];
            }
#pragma unroll
            for (int ti = 0; ti < 9; ++ti) {
                acc[ti] = __builtin_amdgcn_wmma_f32_16x16x4_f32(
                    false, a, false, bf[ti], (short)0, acc[ti], false, false);
            }
        }
    }

    // Store C tiles: VGPR v holds row (v + 8*hi), col = laneM within tile
#pragma unroll
    for (int ti = 0; ti < 9; ++ti) {
        const int l1 = c_tileL1[ti];
        const int d1 = 2 * l1 + 1;
        const int q = l1 * 3 + l2;
        const int ld = TT * d1;
        float* tp = tmpAll + (size_t)b * TMP_PER_B + c_tmpOffQ[q];
        const int col = c_tileColLocal[ti] + laneM;
#pragma unroll
        for (int v = 0; v < 8; ++v) {
            const int row = rowBase + v + 8 * hi;
            tp[(size_t)row * ld + col] = acc[ti][v];
        }
    }
}

// ---------------------------------------------------------------------------
// Build cat_l[tau][256*dl] for one batch: aggregate (CG x tmp), identity, power.
// ---------------------------------------------------------------------------
__launch_bounds__(256)
__global__ void cat_kernel(const float* __restrict__ n0, const float* __restrict__ n1,
                           const float* __restrict__ n2, const float* __restrict__ cgw,
                           const float* __restrict__ tmpAll, float* __restrict__ cat, int b) {
    const int l = blockIdx.y;
    const int dl = 2 * l + 1;
    const int ldn = NN * dl;
    const int tau = (l == 0) ? 976 : 1936;
    const long idx = (long)blockIdx.x * 256 + threadIdx.x;
    if (idx >= (long)tau * ldn) return;
    const int row = (int)(idx / ldn);
    const int col = (int)(idx % ldn);
    const int i = col / dl, m = col % dl;
    const int np = c_npair[l];

    float val = 0.0f;
    if (row < np * 64) {
        // aggregate: row = p*64 + t*4 + s
        const int p = row >> 6, rr = row & 63, t = rr >> 2, s = rr & 3;
        const int e = c_entBase[l] + p;
        const int l1 = c_entL1[e], l2 = c_entL2[e];
        const int d1 = 2 * l1 + 1, d2 = 2 * l2 + 1;
        const float* cg = cgw + c_cgOff[e];
        const float* tp = tmpAll + (size_t)b * TMP_PER_B + c_tmpOffQ[l1 * 3 + l2];
        const int ld = TT * d1;
        const int rbase = (i * SS + s) * d2;
        for (int x = 0; x < d1; ++x)
            for (int y = 0; y < d2; ++y) {
                const float cc = cg[(x * d2 + y) * dl + m];
                if (cc != 0.0f)
                    val += cc * tp[(size_t)(rbase + y) * ld + t * d1 + x];
            }
    } else if (row < np * 64 + TT) {
        // identity (unmasked node; mask applied at output)
        const int t = row - np * 64;
        const float* nl = (l == 0) ? n0 : ((l == 1) ? n1 : n2);
        val = nl[((size_t)(b * NN + i) * TT + t) * dl + m];
    } else {
        // power: row = base + p*256 + t*16 + s
        const int rr = row - np * 64 - TT;
        const int p = rr >> 8, t = (rr >> 4) & 15, s = rr & 15;
        const int e = c_entBase[l] + p;
        const int l1 = c_entL1[e], l2 = c_entL2[e];
        const int d1 = 2 * l1 + 1, d2 = 2 * l2 + 1;
        const float* cg = cgw + c_cgOff[e];
        const float* na = ((l1 == 0) ? n0 : ((l1 == 1) ? n1 : n2)) +
                          ((size_t)(b * NN + i) * TT + t) * d1;
        const float* nb = ((l2 == 0) ? n0 : ((l2 == 1) ? n1 : n2)) +
                          ((size_t)(b * NN + i) * TT + s) * d2;
        for (int x = 0; x < d1; ++x)
            for (int y = 0; y < d2; ++y) {
                const float cc = cg[(x * d2 + y) * dl + m];
                if (cc != 0.0f) val += cc * na[x] * nb[y];
            }
    }
    cat[(size_t)row * ldn + col + ((l == 0) ? 0 : ((l == 1) ? 249856 : 1736704))] = val;
}

// ---------------------------------------------------------------------------
// CatMix GEMM: out[c,(i,m)] = W_l(16 x tau) x cat(tau x 256*dl), masked store.
// W chunk cached in padded LDS (pitch 68 -> conflict-free row reads).
// grid = 2*dl blocks of 8 waves; each wave owns one 16x16 C tile.
// ---------------------------------------------------------------------------
__launch_bounds__(256)
__global__ void mix_kernel(const float* __restrict__ W, const float* __restrict__ cat,
                           const int* __restrict__ mask, float* __restrict__ out,
                           int b, int l) {
    const int dl = 2 * l + 1;
    const int ldn = NN * dl;
    const int tau = (l == 0) ? 976 : 1936;
    const int tid = threadIdx.x;
    const int wave = tid >> 5, lane = tid & 31;
    const int laneM = lane & 15, hi = lane >> 4;
    const int colBase = (blockIdx.x * 8 + wave) * 16;

    __shared__ float Wsh[16 * 68];
    v8f acc = {0, 0, 0, 0, 0, 0, 0, 0};

    for (int k0 = 0; k0 < tau; k0 += 64) {
        const int kw = (tau - k0 < 64) ? (tau - k0) : 64;
        __syncthreads();
        for (int idx = tid; idx < 16 * kw; idx += 256) {
            const int rr = idx / kw, kk = idx % kw;
            Wsh[rr * 68 + kk] = W[(size_t)rr * tau + k0 + kk];
        }
        __syncthreads();
        for (int kk = 0; kk < kw; kk += 4) {
            const int kb = kk + hi * 2;
            v2f a;
            a.x = Wsh[laneM * 68 + kb];
            a.y = Wsh[laneM * 68 + kb + 1];
            v2f bf;
            const float* cp = cat + (size_t)(k0 + kb) * ldn + colBase + laneM;
            bf.x = cp[0];
            bf.y = cp[ldn];
            acc = __builtin_amdgcn_wmma_f32_16x16x4_f32(
                false, a, false, bf, (short)0, acc, false, false);
        }
    }

    const int col = colBase + laneM;
    const int i = col / dl, m = col % dl;
    const float mf = (float)mask[b * NN + i];
#pragma unroll
    for (int v = 0; v < 8; ++v) {
        const int c = v + 8 * hi;
        out[((size_t)(b * NN + i) * NCH + c) * dl + m] = acc[v] * mf;
    }
}

// ---------------------------------------------------------------------------
extern "C" void kernel_launch(void* const* d_in, const int* in_sizes, int n_in,
                              void* d_out, int out_size, void* d_ws, size_t ws_size,
                              hipStream_t stream) {
    (void)in_sizes; (void)n_in; (void)out_size;
    const float* node0 = (const float*)d_in[0];
    const float* edge0 = (const float*)d_in[1];
    const float* W0    = (const float*)d_in[2];
    const float* node1 = (const float*)d_in[3];
    const float* edge1 = (const float*)d_in[4];
    const float* W1    = (const float*)d_in[5];
    const float* node2 = (const float*)d_in[6];
    const float* edge2 = (const float*)d_in[7];
    const float* W2    = (const float*)d_in[8];
    const int*   mask  = (const int*)d_in[9];
    float* out = (float*)d_out;
    float* ws  = (float*)d_ws;

    // workspace layout (floats)
    const size_t WS_CG  = 0;                       // 615  (padded to 1024)
    const size_t WS_NM  = 1024;                    // 8 * 36864
    const size_t WS_TMP = WS_NM + 8 * 36864;       // 8 * 1327104
    const size_t WS_CAT = WS_TMP + (size_t)8 * TMP_PER_B;  // 4214784
    const size_t WS_END = WS_CAT + 4214784;        // ~15.13M floats ~= 60.5 MB
    if (ws_size < WS_END * sizeof(float)) return;

    float* cgw = ws + WS_CG;
    float* nm  = ws + WS_NM;
    float* tmp = ws + WS_TMP;
    float* cat = ws + WS_CAT;

    cg_kernel<<<3, 256, 0, stream>>>(cgw);
    mask_node_kernel<<<dim3(144, 8), 256, 0, stream>>>(node0, node1, node2, mask, nm);

    const float* edges[3] = {edge0, edge1, edge2};
    for (int l2 = 0; l2 < 3; ++l2)
        agg_gemm_kernel<<<dim3(8 * (2 * l2 + 1), 8), 256, 0, stream>>>(edges[l2], nm, tmp, l2);

    const float* Wl[3]     = {W0, W1, W2};
    const size_t catOff[3] = {0, 249856, 1736704};
    const size_t outOff[3] = {0, 32768, 131072};
    for (int b = 0; b < 8; ++b) {
        cat_kernel<<<dim3(9680, 3), 256, 0, stream>>>(node0, node1, node2, cgw, tmp, cat, b);
        for (int l = 0; l < 3; ++l)
            mix_kernel<<<dim3(2 * (2 * l + 1)), 256, 0, stream>>>(
                Wl[l], cat + catOff[l], mask, out + outOff[l], b, l);
    }
}